// NodeConsensusVector_27281632264460
// MI455X (gfx1250) — compile-verified
//
#include <hip/hip_runtime.h>
#include <math.h>

#define N   2048
#define NA  100
#define FWB 64
#define NB  32   // N / FWB

typedef __attribute__((ext_vector_type(2))) float v2f;
typedef __attribute__((ext_vector_type(8))) float v8f;

// 1) w[i][j] = min(f(mg[i][j]*nw[i]), f(mg[j][i]*nw[j])), f(x)=x>0?x:inf, diag=0
__global__ void k_build_w(const float* __restrict__ mg, const float* __restrict__ nw,
                          float* __restrict__ dist) {
  int idx = blockIdx.x * 256 + threadIdx.x;
  int i = idx >> 11, j = idx & (N - 1);
  float aij = mg[i * N + j] * nw[i];
  float aji = mg[j * N + i] * nw[j];
  float wij = aij > 0.0f ? aij : INFINITY;
  float wji = aji > 0.0f ? aji : INFINITY;
  float w = fminf(wij, wji);
  if (i == j) w = 0.0f;
  dist[idx] = w;
}

// 2) Blocked Floyd-Warshall (tile 64x64). Non-negative weights => row/col k
//    are invariant during step k, so in-place parallel updates are safe.
__global__ void k_fw_phase1(float* __restrict__ dist, int kb) {
  __shared__ float T[FWB][FWB + 1];
  const int t = threadIdx.x;
  const long base = (long)(kb * FWB) * N + kb * FWB;
#pragma unroll
  for (int s = 0; s < 16; ++s) {
    int e = t + 256 * s;
    T[e >> 6][e & 63] = dist[base + (long)(e >> 6) * N + (e & 63)];
  }
  __syncthreads();
  const int row = t >> 2;
  const int c0 = (t & 3) * 16;
  for (int kk = 0; kk < FWB; ++kk) {
    if (row != kk) {
      float dk = T[row][kk];
#pragma unroll
      for (int cc = 0; cc < 16; ++cc) {
        int c = c0 + cc;
        if (c != kk) T[row][c] = fminf(T[row][c], dk + T[kk][c]);
      }
    }
    __syncthreads();
  }
#pragma unroll
  for (int s = 0; s < 16; ++s) {
    int e = t + 256 * s;
    dist[base + (long)(e >> 6) * N + (e & 63)] = T[e >> 6][e & 63];
  }
}

__global__ void k_fw_phase2(float* __restrict__ dist, int kb) {
  int b = blockIdx.x;
  bool isRow = b < NB;
  int o = b & (NB - 1);
  if (o == kb) return;  // uniform per-block exit
  __shared__ float D[FWB][FWB + 1];
  __shared__ float C[FWB][FWB + 1];
  const int t = threadIdx.x;
  const long dbase = (long)(kb * FWB) * N + kb * FWB;
  const long cbase = isRow ? ((long)(kb * FWB) * N + o * FWB)
                           : ((long)(o * FWB) * N + kb * FWB);
#pragma unroll
  for (int s = 0; s < 16; ++s) {
    int e = t + 256 * s; int r = e >> 6, c = e & 63;
    D[r][c] = dist[dbase + (long)r * N + c];
    C[r][c] = dist[cbase + (long)r * N + c];
  }
  __syncthreads();
  const int row = t >> 2;
  const int c0 = (t & 3) * 16;
  if (isRow) {
    for (int kk = 0; kk < FWB; ++kk) {         // C(kb,o): uses C's own row kk
      if (row != kk) {
        float dk = D[row][kk];
#pragma unroll
        for (int cc = 0; cc < 16; ++cc) {
          int c = c0 + cc;
          C[row][c] = fminf(C[row][c], dk + C[kk][c]);
        }
      }
      __syncthreads();
    }
  } else {
    for (int kk = 0; kk < FWB; ++kk) {         // C(o,kb): uses C's own col kk
      float ck = C[row][kk];
#pragma unroll
      for (int cc = 0; cc < 16; ++cc) {
        int c = c0 + cc;
        if (c != kk) C[row][c] = fminf(C[row][c], ck + D[kk][c]);
      }
      __syncthreads();
    }
  }
#pragma unroll
  for (int s = 0; s < 16; ++s) {
    int e = t + 256 * s;
    dist[cbase + (long)(e >> 6) * N + (e & 63)] = C[e >> 6][e & 63];
  }
}

__global__ void k_fw_phase3(float* __restrict__ dist, int kb) {
  int bi = blockIdx.x, bj = blockIdx.y;
  if (bi == kb || bj == kb) return;  // uniform per-block exit
  __shared__ float A[FWB][FWB + 1];
  __shared__ float Bs[FWB][FWB + 1];
  const int t = threadIdx.x;
  const long abase = (long)(bi * FWB) * N + kb * FWB;
  const long bbase = (long)(kb * FWB) * N + bj * FWB;
  const long cbase = (long)(bi * FWB) * N + bj * FWB;
#pragma unroll
  for (int s = 0; s < 16; ++s) {
    int e = t + 256 * s; int r = e >> 6, c = e & 63;
    A[r][c]  = dist[abase + (long)r * N + c];
    Bs[r][c] = dist[bbase + (long)r * N + c];
  }
  const int r0 = (t >> 4) * 4;
  const int c0 = (t & 15) * 4;
  float creg[4][4];
#pragma unroll
  for (int dr = 0; dr < 4; ++dr)
#pragma unroll
    for (int dc = 0; dc < 4; ++dc)
      creg[dr][dc] = dist[cbase + (long)(r0 + dr) * N + (c0 + dc)];
  __syncthreads();
#pragma unroll 4
  for (int kk = 0; kk < FWB; ++kk) {
    float a[4], b[4];
#pragma unroll
    for (int dr = 0; dr < 4; ++dr) a[dr] = A[r0 + dr][kk];
#pragma unroll
    for (int dc = 0; dc < 4; ++dc) b[dc] = Bs[kk][c0 + dc];
#pragma unroll
    for (int dr = 0; dr < 4; ++dr)
#pragma unroll
      for (int dc = 0; dc < 4; ++dc)
        creg[dr][dc] = fminf(creg[dr][dc], a[dr] + b[dc]);
  }
#pragma unroll
  for (int dr = 0; dr < 4; ++dr)
#pragma unroll
    for (int dc = 0; dc < 4; ++dc)
      dist[cbase + (long)(r0 + dr) * N + (c0 + dc)] = creg[dr][dc];
}

// 3) anchors = argsort(-node_weights)[:100], stable ties -> lower index
__global__ void k_anchors(const float* __restrict__ nw, int* __restrict__ anchors) {
  int i = blockIdx.x * 256 + threadIdx.x;
  float wi = nw[i];
  int rank = 0;
  for (int j = 0; j < N; ++j) {
    float wj = nw[j];
    rank += ((wj > wi) || ((wj == wi) && (j < i))) ? 1 : 0;
  }
  if (rank < NA) anchors[rank] = i;
}

// 4) consensus_vectors gather (+inf -> 100) and weighted = wm * cv
__global__ void k_cv(const float* __restrict__ dist, const int* __restrict__ anchors,
                     const float* __restrict__ wm, float* __restrict__ cv,
                     float* __restrict__ weighted) {
  int idx = blockIdx.x * 256 + threadIdx.x;
  int i = idx / NA, a = idx - i * NA;
  float v = dist[(long)i * N + anchors[a]];
  if (isinf(v)) v = 100.0f;
  cv[idx] = v;
  weighted[idx] = wm[idx] * v;
}

// 5) (sum(cv) - 100*N) / (N*NA), deterministic fixed-order reduction
__global__ void k_dsum(const float* __restrict__ cv, float* __restrict__ outp) {
  __shared__ float red[256];
  int t = threadIdx.x;
  float s = 0.0f;
  for (int i = t; i < N * NA; i += 256) s += cv[i];
  red[t] = s;
  __syncthreads();
  for (int off = 128; off > 0; off >>= 1) {
    if (t < off) red[t] += red[t + off];
    __syncthreads();
  }
  if (t == 0) outp[0] = (red[0] - 100.0f * (float)N) / ((float)N * (float)NA);
}

// 6) xx[i] = sum_a weighted[i][a]^2
__global__ void k_xx(const float* __restrict__ weighted, float* __restrict__ xx) {
  int i = blockIdx.x * 256 + threadIdx.x;
  const float* row = weighted + i * NA;
  float s = 0.0f;
  for (int a = 0; a < NA; ++a) { float v = row[a]; s += v * v; }
  xx[i] = s;
}

// 7) Gram = W @ W^T via V_WMMA_F32_16X16X4_F32 (exact fp32) with fused
//    Gaussian-kernel epilogue. One full wave (EXEC all-ones) per 16x16 tile.
__global__ void k_gram(const float* __restrict__ W, const float* __restrict__ xx,
                       const float* __restrict__ sigma, float* __restrict__ cg) {
  const int tile = blockIdx.x;
  const int m0 = (tile >> 7) * 16;
  const int n0 = (tile & 127) * 16;
  const int lane = threadIdx.x;
  const int lr = lane & 15;
  const int kOff = (lane >> 4) << 1;   // lanes 0-15: K pair {0,1}; 16-31: {2,3}
  const float* arow = W + (m0 + lr) * NA + kOff;
  const float* brow = W + (n0 + lr) * NA + kOff;
  v8f acc = {0.f, 0.f, 0.f, 0.f, 0.f, 0.f, 0.f, 0.f};
#pragma unroll
  for (int s = 0; s < 25; ++s) {       // K = 100 = 25 steps of 4
    v2f av = *(const v2f*)(arow + 4 * s);
    v2f bv = *(const v2f*)(brow + 4 * s);
    acc = __builtin_amdgcn_wmma_f32_16x16x4_f32(false, av, false, bv,
                                                (short)0, acc, false, false);
  }
  const float sg = *sigma;
  const float inv = 1.0f / (2.0f * sg * sg);
  const int col = n0 + lr;
  const float xc = xx[col];
  const int rbase = m0 + ((lane >> 4) << 3);  // C/D layout: VGPR r -> M = r + 8*(lane/16)
#pragma unroll
  for (int r = 0; r < 8; ++r) {
    int row = rbase + r;
    float sq = fmaxf(xx[row] + xc - 2.0f * acc[r], 0.0f);
    cg[(long)row * N + col] = expf(-(sq * sq) * inv);
  }
}

extern "C" void kernel_launch(void* const* d_in, const int* in_sizes, int n_in,
                              void* d_out, int out_size, void* d_ws, size_t ws_size,
                              hipStream_t stream) {
  const float* mg    = (const float*)d_in[0];
  const float* nw    = (const float*)d_in[1];
  const float* wm    = (const float*)d_in[2];
  const float* sigma = (const float*)d_in[3];

  float* out      = (float*)d_out;
  float* cv_out   = out;                 // N*NA floats
  float* dsum_out = out + N * NA;        // 1 float
  float* cg       = out + N * NA + 1;    // N*N floats
  float* dist     = cg;                  // reuse output graph region as FW scratch

  float* weighted = (float*)d_ws;        // N*NA floats (~800KB total ws use)
  float* xx       = weighted + N * NA;   // N floats
  int*   anchors  = (int*)(xx + N);      // NA ints

  k_build_w<<<(N * N) / 256, 256, 0, stream>>>(mg, nw, dist);
  for (int kb = 0; kb < NB; ++kb) {
    k_fw_phase1<<<1, 256, 0, stream>>>(dist, kb);
    k_fw_phase2<<<2 * NB, 256, 0, stream>>>(dist, kb);
    k_fw_phase3<<<dim3(NB, NB), 256, 0, stream>>>(dist, kb);
  }
  k_anchors<<<N / 256, 256, 0, stream>>>(nw, anchors);
  k_cv<<<(N * NA) / 256, 256, 0, stream>>>(dist, anchors, wm, cv_out, weighted);
  k_dsum<<<1, 256, 0, stream>>>(cv_out, dsum_out);
  k_xx<<<N / 256, 256, 0, stream>>>(weighted, xx);
  // cg aliases dist; safe to overwrite now that k_cv has consumed dist
  k_gram<<<(N / 16) * (N / 16), 32, 0, stream>>>(weighted, xx, sigma, cg);
}